// TwoTowerModel_42855183680013
// MI455X (gfx1250) — compile-verified
//
#include <hip/hip_runtime.h>
#include <hip/hip_bf16.h>

// ---------------------------------------------------------------------------
// Two-tower retrieval: Q[512,64] x items^T[64,500000] -> top-100 per query.
// Phase 1: WMMA f32 16x16x4 GEMM into LDS + per-chunk radix-select top-100.
// Phase 2: per-query merge of 245 partial lists + bitonic sort of final 100.
// ---------------------------------------------------------------------------

#define BQ      512
#define NITEMS  500000
#define DIM     64
#define TOPK    100
#define CHUNK   2048
#define NCHUNK  ((NITEMS + CHUNK - 1) / CHUNK)   // 245
#define QT      16
#define NQT     (BQ / QT)                        // 32
#define SLD     (CHUNK + 4)                      // padded LDS row stride (floats)
#define CAND    (NCHUNK * TOPK)                  // 24500 candidates per query

typedef __attribute__((ext_vector_type(2))) float v2f;
typedef __attribute__((ext_vector_type(8))) float v8f;

__device__ __forceinline__ v8f wmma_f32_k4(v2f a, v2f b, v8f c) {
  return __builtin_amdgcn_wmma_f32_16x16x4_f32(false, a, false, b,
                                               (short)0, c, false, false);
}

// Order-preserving float -> uint key (bigger key == bigger float), exact inverse.
__device__ __forceinline__ unsigned f2key(float f) {
  unsigned u = __float_as_uint(f);
  return (u & 0x80000000u) ? ~u : (u | 0x80000000u);
}
__device__ __forceinline__ float key2f(unsigned k) {
  return __uint_as_float((k & 0x80000000u) ? (k ^ 0x80000000u) : ~k);
}

// "a ranks before b" in descending score order, ties broken by lower id
// (matches jax.lax.top_k stability).
__device__ __forceinline__ bool better(float2 a, float2 b) {
  unsigned ka = f2key(a.x), kb = f2key(b.x);
  return (ka > kb) || (ka == kb && a.y < b.y);
}

// ---------------------------------------------------------------------------
// Phase 1: one block = 16 queries x 2048 items. WMMA scores into LDS, then
// per-row bitwise radix select of the top-100, emitted to workspace.
// ---------------------------------------------------------------------------
__global__ __launch_bounds__(256) void tt_gemm_partk(
    const int* __restrict__ uid, const float* __restrict__ utab,
    const float* __restrict__ items, float2* __restrict__ part) {
  __shared__ float qs[QT * DIM];        // 4 KB   query tile
  __shared__ float sc[QT * SLD];        // 131 KB score tile (reused as keys)
  __shared__ int   cnt_gt[QT];
  __shared__ int   cnt_eq[QT];

  const int qtile = blockIdx.x;
  const int chunk = blockIdx.y;
  const int tid   = threadIdx.x;

  // Gather the 16 query embeddings (user tower lookup).
  for (int j = tid; j < QT * DIM; j += 256) {
    int m = j >> 6, d = j & 63;
    qs[j] = utab[(size_t)uid[qtile * QT + m] * DIM + d];
  }
  __syncthreads();

  const int lane = tid & 31;
  const int wave = tid >> 5;
  const int lm   = lane & 15;   // M (A) / N (B,C) index
  const int lh   = lane >> 4;   // half-wave selector

  // Preload A fragments: 16x4 f32 layout — lanes 0-15: K0,K1; lanes 16-31: K2,K3.
  v2f a[16];
#pragma unroll
  for (int k = 0; k < 16; ++k) {
    a[k].x = qs[lm * DIM + 4 * k + 2 * lh + 0];
    a[k].y = qs[lm * DIM + 4 * k + 2 * lh + 1];
  }

  const int cbase  = chunk * CHUNK;
  const int nvalid = (NITEMS - cbase) < CHUNK ? (NITEMS - cbase) : CHUNK;
  const int tmax   = nvalid >> 4;       // fully-valid 16-item tiles (NITEMS%16==0)

  const float* ibase = items + 2 * lh;  // per-lane dim offset within a row

  // 128 item tiles of 16; each wave owns tiles {wave, wave+8, ...}.
  // Main loop: two tiles in flight (independent WMMA chains), no guards.
  int t = wave;
  for (; t + 8 < tmax; t += 16) {
    const v2f* r0 = (const v2f*)(ibase + (size_t)(cbase + t * 16 + lm) * DIM);
    const v2f* r1 = (const v2f*)(ibase + (size_t)(cbase + (t + 8) * 16 + lm) * DIM);
    v8f acc0 = {};
    v8f acc1 = {};
#pragma unroll
    for (int k = 0; k < 16; ++k) {
      v2f b0 = r0[2 * k];               // global_load_b64, dims 4k..4k+1 (+2*lh)
      v2f b1 = r1[2 * k];
      acc0 = wmma_f32_k4(a[k], b0, acc0);
      acc1 = wmma_f32_k4(a[k], b1, acc1);
    }
    // C/D layout: VGPR r, lane -> M = r + 8*lh, N = lm.
#pragma unroll
    for (int r = 0; r < 8; ++r) {
      const int m = r + 8 * lh;
      sc[m * SLD + t * 16 + lm]       = acc0[r];
      sc[m * SLD + (t + 8) * 16 + lm] = acc1[r];
    }
  }
  // Remainder (only in the final partial chunk).
  for (; t < tmax; t += 8) {
    const v2f* r0 = (const v2f*)(ibase + (size_t)(cbase + t * 16 + lm) * DIM);
    v8f acc0 = {};
#pragma unroll
    for (int k = 0; k < 16; ++k) acc0 = wmma_f32_k4(a[k], r0[2 * k], acc0);
#pragma unroll
    for (int r = 0; r < 8; ++r)
      sc[(r + 8 * lh) * SLD + t * 16 + lm] = acc0[r];
  }
  // Pad the invalid tail (items >= NITEMS) with -inf so selection ignores it.
  if (tmax < CHUNK / 16) {
    for (int m = 0; m < QT; ++m)
      for (int c = (tmax << 4) + tid; c < CHUNK; c += 256)
        sc[m * SLD + c] = -__builtin_inff();
  }
  __syncthreads();

  // --- per-row top-100 selection: 16 threads per row, all in one wave half ---
  const int row = tid >> 4;
  const int sub = tid & 15;
  unsigned* krow = reinterpret_cast<unsigned*>(&sc[row * SLD]);

  // Convert scores to monotone keys in place (group-private, same wave).
  for (int e = sub; e < CHUNK; e += 16) krow[e] = f2key(sc[row * SLD + e]);

  // Bitwise radix search for the 100th-largest key.
  unsigned thr = 0u;
  for (int bit = 31; bit >= 0; --bit) {
    const unsigned candk = thr | (1u << bit);
    int c = 0;
    for (int e = sub; e < CHUNK; e += 16) c += (krow[e] >= candk) ? 1 : 0;
    c += __shfl_xor(c, 1, 16);
    c += __shfl_xor(c, 2, 16);
    c += __shfl_xor(c, 4, 16);
    c += __shfl_xor(c, 8, 16);
    if (c >= TOPK) thr = candk;
  }

  // Count strictly-greater entries.
  int cg = 0;
  for (int e = sub; e < CHUNK; e += 16) cg += (krow[e] > thr) ? 1 : 0;
  cg += __shfl_xor(cg, 1, 16);
  cg += __shfl_xor(cg, 2, 16);
  cg += __shfl_xor(cg, 4, 16);
  cg += __shfl_xor(cg, 8, 16);

  if (sub == 0) { cnt_gt[row] = 0; cnt_eq[row] = 0; }  // same wave: LDS in order

  float2* dst = part + ((size_t)(qtile * QT + row) * NCHUNK + chunk) * TOPK;
  for (int e = sub; e < CHUNK; e += 16) {
    const unsigned kk = krow[e];
    if (kk > thr) {
      const int p = atomicAdd(&cnt_gt[row], 1);
      dst[p] = make_float2(key2f(kk), (float)(cbase + e));
    }
  }
  for (int e = sub; e < CHUNK; e += 16) {
    if (krow[e] == thr) {
      const int p = atomicAdd(&cnt_eq[row], 1);
      const int slot = cg + p;
      if (slot < TOPK) dst[slot] = make_float2(key2f(thr), (float)(cbase + e));
    }
  }
}

// ---------------------------------------------------------------------------
// Phase 2: one block per query. Merge 245 partial top-100 lists (24500
// candidates), radix-select the global top-100, bitonic sort, write out.
// ---------------------------------------------------------------------------
__global__ __launch_bounds__(256) void tt_merge_topk(
    const float2* __restrict__ part, float* __restrict__ out) {
  __shared__ unsigned keys[CAND];   // 98 KB
  __shared__ float2   buf[128];
  __shared__ int      wsum[8];
  __shared__ int      bc;
  __shared__ int      cgt_s, ctr_gt, ctr_eq;

  const int q    = blockIdx.x;
  const int tid  = threadIdx.x;
  const int lane = tid & 31;
  const int wv   = tid >> 5;
  const float2* src = part + (size_t)q * CAND;

  for (int j = tid; j < CAND; j += 256) keys[j] = f2key(src[j].x);
  __syncthreads();

  unsigned thr = 0u;
  for (int bit = 31; bit >= 0; --bit) {
    const unsigned candk = thr | (1u << bit);
    int c = 0;
    for (int j = tid; j < CAND; j += 256) c += (keys[j] >= candk) ? 1 : 0;
    c += __shfl_xor(c, 1);
    c += __shfl_xor(c, 2);
    c += __shfl_xor(c, 4);
    c += __shfl_xor(c, 8);
    c += __shfl_xor(c, 16);
    if (lane == 0) wsum[wv] = c;
    __syncthreads();
    if (tid == 0) {
      int t = 0;
#pragma unroll
      for (int w = 0; w < 8; ++w) t += wsum[w];
      bc = t;
    }
    __syncthreads();
    if (bc >= TOPK) thr = candk;
  }

  int cg = 0;
  for (int j = tid; j < CAND; j += 256) cg += (keys[j] > thr) ? 1 : 0;
  cg += __shfl_xor(cg, 1);
  cg += __shfl_xor(cg, 2);
  cg += __shfl_xor(cg, 4);
  cg += __shfl_xor(cg, 8);
  cg += __shfl_xor(cg, 16);
  if (lane == 0) wsum[wv] = cg;
  __syncthreads();
  if (tid == 0) {
    int t = 0;
#pragma unroll
    for (int w = 0; w < 8; ++w) t += wsum[w];
    cgt_s = t;
    ctr_gt = 0;
    ctr_eq = 0;
  }
  for (int j = tid; j < 128; j += 256)
    buf[j] = make_float2(-__builtin_inff(), 3.0e8f);   // pad slots
  __syncthreads();

  for (int j = tid; j < CAND; j += 256) {
    if (keys[j] > thr) {
      const int p = atomicAdd(&ctr_gt, 1);
      buf[p] = src[j];
    }
  }
  __syncthreads();
  const int cgt = cgt_s;
  for (int j = tid; j < CAND; j += 256) {
    if (keys[j] == thr) {
      const int p = atomicAdd(&ctr_eq, 1);
      if (cgt + p < TOPK) buf[cgt + p] = src[j];
    }
  }

  // Bitonic sort 128 entries, descending (score desc, id asc).
  for (int ksz = 2; ksz <= 128; ksz <<= 1) {
    for (int jsz = ksz >> 1; jsz > 0; jsz >>= 1) {
      __syncthreads();
      if (tid < 128) {
        const int i = tid;
        const int p = i ^ jsz;
        if (p > i) {
          float2 A  = buf[i];
          float2 Bv = buf[p];
          const bool desc   = ((i & ksz) == 0);
          const bool doSwap = desc ? better(Bv, A) : better(A, Bv);
          if (doSwap) { buf[i] = Bv; buf[p] = A; }
        }
      }
    }
  }
  __syncthreads();

  if (tid < TOPK) {
    out[(size_t)q * TOPK + tid] = buf[tid].x;                         // scores
    out[(size_t)BQ * TOPK + (size_t)q * TOPK + tid] = buf[tid].y;     // ids
  }
}

// ---------------------------------------------------------------------------
extern "C" void kernel_launch(void* const* d_in, const int* in_sizes, int n_in,
                              void* d_out, int out_size, void* d_ws, size_t ws_size,
                              hipStream_t stream) {
  const int*   uid   = (const int*)d_in[0];     // user_ids [512] int32
  const float* utab  = (const float*)d_in[1];   // user_table [1000002,64] f32
  const float* items = (const float*)d_in[2];   // item_embs [500000,64] f32
  // d_in[3] is k (==100), compiled in as TOPK.

  float2* part = (float2*)d_ws;                 // [512][245][100] (score,id) ~100.4 MB
  float*  out  = (float*)d_out;                 // [512*100] scores ++ [512*100] ids

  dim3 g1(NQT, NCHUNK);
  tt_gemm_partk<<<g1, 256, 0, stream>>>(uid, utab, items, part);
  tt_merge_topk<<<BQ, 256, 0, stream>>>(part, out);
}